// ProductQuantization_11192684773820
// MI455X (gfx1250) — compile-verified
//
#include <hip/hip_runtime.h>

// ---------------------------------------------------------------------------
// Product quantization + EMA codebook update for MI455X (gfx1250, wave32).
//
// x:     [B=2, L=4096, H=16, D=64] f32  -> tokens T = 8192 per head
// means: [H=16, K=1024, D=64] f32
// out:   cluster_ids [B,L,H] (int32, 131072 elems) ++ new_means [H,K,D] (f32)
//
// - Per-head codebook (256 KB) DMA'd into LDS by the Tensor Data Mover with
//   pad_interval=64 dwords / pad_amount=4 dwords -> lands directly in the
//   bank-conflict-free 68-float row stride the WMMA B loads use.
// - fp32 WMMA 16x16x4 for the distance GEMM (exact fp32 -> argmin-safe),
//   double-buffered LDS B operands so ds waits never sit on the WMMA chain.
// - shfl_xor argmin reduction, atomic scatter-add for the EMA statistics.
// ---------------------------------------------------------------------------

typedef float v2f __attribute__((ext_vector_type(2)));
typedef float v4f __attribute__((ext_vector_type(4)));
typedef float v8f __attribute__((ext_vector_type(8)));
typedef unsigned int v4u __attribute__((ext_vector_type(4)));
typedef int v4i_t __attribute__((ext_vector_type(4)));
typedef int v8i_t __attribute__((ext_vector_type(8)));

#define NHEAD 16
#define NCLUST 1024
#define DIM 64
#define NTOK 8192            // B*L
#define ROWSTRIDE 68         // padded LDS row stride in floats (bank-conflict-free)
#define TOK_PER_BLOCK 128
#define NWAVE 8
#define TILES_PER_HEAD (NTOK / TOK_PER_BLOCK)   // 64

// LDS layout in floats
#define LDS_MEANS 0
#define LDS_MNORM (NCLUST * ROWSTRIDE)                    // 69632
#define LDS_IDS   (NCLUST * ROWSTRIDE + NCLUST)           // 70656
#define LDS_FLOATS (NCLUST * ROWSTRIDE + NCLUST + NWAVE * 16)  // 70784 -> 283,136 B

extern "C" __global__ void pq_zero_kernel(float* __restrict__ p, int n) {
    int i = blockIdx.x * blockDim.x + threadIdx.x;
    if (i < n) p[i] = 0.0f;
}

// Issue one TDM descriptor: copy [NCLUST x DIM] f32 tile (row-major, stride
// DIM) from `gsrc` into LDS at byte offset `lds_byte`, padding 4 dwords after
// every 64 dwords (-> ROWSTRIDE=68 floats effective row pitch in LDS).
__device__ __forceinline__ void tdm_load_codebook(const float* gsrc, unsigned lds_byte) {
    unsigned long long ga = (unsigned long long)gsrc;

    v4u g0;
    g0.x = 1u;                                        // count=1 (valid), user mode
    g0.y = lds_byte;                                  // lds_addr [63:32]
    g0.z = (unsigned)(ga & 0xffffffffu);              // global_addr [95:64]
    g0.w = (unsigned)((ga >> 32) & 0x01ffffffu)       // global_addr [120:96]
         | (2u << 30);                                // type=2 ("image") [127:126]

    v8i_t g1;
    g1[0] = (2 << 16)                                 // data_size = 4 bytes
          | (1 << 20)                                 // pad_enable
          | (5 << 22)                                 // pad_interval: every 64 dwords
          | (3 << 25);                                // pad_amount: 4 dwords
    g1[1] = (DIM & 0xffff) << 16;                     // tensor_dim0[15:0]=64; atomic_barrier_addr=0
    g1[2] = (NCLUST & 0xffff) << 16;                  // tensor_dim0[31:16]=0; tensor_dim1[15:0]=1024
    g1[3] = (DIM & 0xffff) << 16;                     // tensor_dim1[31:16]=0; tile_dim0=64
    g1[4] = (NCLUST & 0xffff);                        // tile_dim1=1024; tile_dim2=0
    g1[5] = DIM;                                      // tensor_dim0_stride[31:0]=64
    g1[6] = 0;                                        // stride0[47:32]=0; stride1[15:0]=0
    g1[7] = 0;                                        // tensor_dim1_stride[47:16]=0

    v4i_t z4 = {0, 0, 0, 0};
#if defined(__clang_major__) && (__clang_major__ >= 23)
    v8i_t z8 = {0, 0, 0, 0, 0, 0, 0, 0};
    __builtin_amdgcn_tensor_load_to_lds(g0, g1, z4, z4, z8, 0);
#else
    __builtin_amdgcn_tensor_load_to_lds(g0, g1, z4, z4, 0);
#endif
}

extern "C" __global__ __launch_bounds__(256, 1)
void pq_assign_kernel(const float* __restrict__ x,
                      const float* __restrict__ means,
                      int* __restrict__ out_ids,
                      float* __restrict__ counts,
                      float* __restrict__ sums)
{
    extern __shared__ float lds[];
    const int tid  = threadIdx.x;
    const int bid  = blockIdx.x;
    const int h    = bid >> 6;        // 64 tiles per head
    const int tile = bid & 63;
    const int wave = tid >> 5;
    const int lane = tid & 31;
    const int r    = lane & 15;       // row-within-halfwave / column index
    const int hh   = lane >> 4;       // half-wave select

    // ---- async-DMA this head's codebook into padded LDS via the TDM ----
    const float* mh = means + (size_t)h * (NCLUST * DIM);
    if (wave == 0) {
        // dynamic LDS segment starts right after static LDS
        tdm_load_codebook(mh, __builtin_amdgcn_groupstaticsize());
    }

    // ---- overlap: preload the wave's 16x64 A tile (WMMA f32 16x16x4 layout)
    // lane<16 : a[kk] = X[row][4kk+0], X[row][4kk+1]
    // lane>=16: a[kk] = X[row][4kk+2], X[row][4kk+3]
    const int trow = tile * TOK_PER_BLOCK + wave * 16 + r;
    const float* xrow = x + (size_t)trow * (NHEAD * DIM) + h * DIM + 2 * hh;
    v2f a[16];
    #pragma unroll
    for (int kk = 0; kk < 16; ++kk) a[kk] = *(const v2f*)(xrow + 4 * kk);

    if (wave == 0) __builtin_amdgcn_s_wait_tensorcnt(0);
    __syncthreads();   // codebook now resident in LDS for all 8 waves

    // ---- per-cluster squared norms ----
    for (int c = tid; c < NCLUST; c += 256) {
        const float* rowp = &lds[LDS_MEANS + c * ROWSTRIDE];
        float s = 0.0f;
        #pragma unroll
        for (int d0 = 0; d0 < DIM; ++d0) s = fmaf(rowp[d0], rowp[d0], s);
        lds[LDS_MNORM + c] = s;
    }
    __syncthreads();

    float best[8];
    int   bcl[8];
    #pragma unroll
    for (int j = 0; j < 8; ++j) { best[j] = 3.4e38f; bcl[j] = 0; }

    // ---- double-buffered sweep of 64 cluster tiles, 16 fp32 WMMAs each ----
    const float* bbase0 = &lds[LDS_MEANS + r * ROWSTRIDE + 2 * hh];

    v2f   b0[16], b1[16];
    float mn0, mn1;

    // prefetch batch for one tile: 8x ds_load_2addr_b64 + 1x ds_load_b32
    #define LOAD_B(BUF, MN, CT)                                              \
        do {                                                                 \
            const float* _bb = bbase0 + (CT) * (16 * ROWSTRIDE);             \
            _Pragma("unroll")                                                \
            for (int kk = 0; kk < 16; ++kk)                                  \
                (BUF)[kk] = *(const v2f*)(_bb + 4 * kk);                     \
            (MN) = lds[LDS_MNORM + (CT) * 16 + r];                           \
        } while (0)

    // consume one tile (no LDS access inside -> no dscnt stall on the chain)
    #define DO_TILE(BUF, MN, CT)                                             \
        do {                                                                 \
            v8f acc = {};                                                    \
            _Pragma("unroll")                                                \
            for (int kk = 0; kk < 16; ++kk)                                  \
                acc = __builtin_amdgcn_wmma_f32_16x16x4_f32(                 \
                    false, a[kk], false, (BUF)[kk], (short)0, acc,           \
                    false, false);                                           \
            const int _c = (CT) * 16 + r;                                    \
            _Pragma("unroll")                                                \
            for (int j = 0; j < 8; ++j) {                                    \
                float _v = fmaf(-2.0f, acc[j], (MN));                        \
                if (_v < best[j]) { best[j] = _v; bcl[j] = _c; }             \
            }                                                                \
        } while (0)

    LOAD_B(b0, mn0, 0);
    for (int ct = 0; ct < NCLUST / 16; ct += 2) {
        LOAD_B(b1, mn1, ct + 1);       // issue next tile's batch
        DO_TILE(b0, mn0, ct);          // waits only on the long-issued b0 batch
        LOAD_B(b0, mn0, ct + 2);       // ct+2==64 reads scratch rows: in-bounds, unused
        DO_TILE(b1, mn1, ct + 1);
    }
    #undef LOAD_B
    #undef DO_TILE

    // ---- argmin reduction across the 16 lanes of each half-wave ----
    #pragma unroll
    for (int off = 1; off <= 8; off <<= 1) {
        #pragma unroll
        for (int j = 0; j < 8; ++j) {
            float ov = __shfl_xor(best[j], off, 32);
            int   oi = __shfl_xor(bcl[j], off, 32);
            if (ov < best[j] || (ov == best[j] && oi < bcl[j])) {
                best[j] = ov; bcl[j] = oi;
            }
        }
    }

    // lanes 0 / 16 own rows 0..7 / 8..15: publish ids (LDS + global)
    if (r == 0) {
        const int mbase = hh * 8;
        #pragma unroll
        for (int j = 0; j < 8; ++j) {
            const int m = mbase + j;
            lds[LDS_IDS + wave * 16 + m] = __int_as_float(bcl[j]);
            const int tok = tile * TOK_PER_BLOCK + wave * 16 + m;
            out_ids[(size_t)tok * NHEAD + h] = bcl[j];
        }
    }
    __syncthreads();

    // ---- EMA statistics scatter: reuse A registers (lane owns 32 dims of its row)
    const int myid = __float_as_int(lds[LDS_IDS + wave * 16 + r]);
    if (hh == 0) atomicAdd(&counts[h * NCLUST + myid], 1.0f);
    float* sb = sums + ((size_t)h * NCLUST + myid) * DIM + 2 * hh;
    #pragma unroll
    for (int kk = 0; kk < 16; ++kk) {
        atomicAdd(&sb[4 * kk + 0], a[kk].x);
        atomicAdd(&sb[4 * kk + 1], a[kk].y);
    }
}

extern "C" __global__ void pq_finalize_kernel(const float* __restrict__ counts,
                                              const float* __restrict__ sums,
                                              const float* __restrict__ means,
                                              float* __restrict__ out_means)
{
    int i = blockIdx.x * blockDim.x + threadIdx.x;
    if (i < NHEAD * NCLUST * DIM) {
        const int hk = i >> 6;            // /DIM
        const float cnt = counts[hk];
        out_means[i] = fmaf(0.999f, means[i], 0.001f * (sums[i] / (1e-6f + cnt)));
    }
}

extern "C" void kernel_launch(void* const* d_in, const int* in_sizes, int n_in,
                              void* d_out, int out_size, void* d_ws, size_t ws_size,
                              hipStream_t stream) {
    const float* x     = (const float*)d_in[0];   // [2,4096,16,64] f32
    const float* means = (const float*)d_in[1];   // [16,1024,64] f32

    int*   out_ids   = (int*)d_out;                       // 131072 int32 ids
    float* out_means = (float*)d_out + (NTOK * NHEAD);    // 1,048,576 f32 means

    float* counts = (float*)d_ws;                 // [16,1024]
    float* sums   = counts + NHEAD * NCLUST;      // [16,1024,64]

    const int nz = NHEAD * NCLUST + NHEAD * NCLUST * DIM;
    pq_zero_kernel<<<(nz + 255) / 256, 256, 0, stream>>>(counts, nz);

    const size_t lds_bytes = (size_t)LDS_FLOATS * sizeof(float);  // 283,136 B
    pq_assign_kernel<<<NHEAD * TILES_PER_HEAD, 256, lds_bytes, stream>>>(
        x, means, out_ids, counts, sums);

    const int nm = NHEAD * NCLUST * DIM;
    pq_finalize_kernel<<<(nm + 255) / 256, 256, 0, stream>>>(
        counts, sums, means, out_means);
}